// DeepEMD_73254962201123
// MI455X (gfx1250) — compile-verified
//
#include <hip/hip_runtime.h>
#include <math.h>

// ---------------------------------------------------------------------------
// DeepEMD logits on MI455X (gfx1250):
//   kernel 1: per-batch stats + WMMA f32 16x16x4 similarity GEMM -> padded ws
//   kernel 2: one wave32 per batch, register-resident log-Sinkhorn (100 it),
//             dual M / M^T register layout, no cross-lane shuffles in loop.
// ---------------------------------------------------------------------------

typedef __attribute__((ext_vector_type(2))) float v2f;
typedef __attribute__((ext_vector_type(8))) float v8f;

constexpr int kC = 512;   // channels
constexpr int kN = 49;    // 7*7 spatial nodes
constexpr int kNP = 64;   // padded tile edge (ws layout)
constexpr float kNegBig = -1.0e30f;

__device__ __forceinline__ float lane_bcast(float x, int lane) {
  return __int_as_float(__builtin_amdgcn_readlane(__float_as_int(x), lane));
}

__device__ __forceinline__ float wave_sum32(float x) {
#pragma unroll
  for (int off = 16; off > 0; off >>= 1)
    x += __shfl_xor(x, off, 32);
  return x;
}

// ---------------------------------------------------------------------------
// Kernel 1: one block (128 threads = 4 waves) per batch element.
//   Phase A: channel means of proto/query (for the weight vectors)
//   Phase B: per-pixel channel-mean, centered L2 norm, weights -> log_r/log_c
//   Phase C: sim = qn . pn^T via V_WMMA_F32_16X16X4_F32, 4x4 tiles of 16x16,
//            stored unconditionally into a padded 64x64 tile (no divergence).
// ---------------------------------------------------------------------------
__global__ __launch_bounds__(128)
void deepemd_sim_kernel(const float* __restrict__ Q, const float* __restrict__ P,
                        float* __restrict__ simw, float* __restrict__ logr,
                        float* __restrict__ logc)
{
  const int b = blockIdx.x;
  const int t = threadIdx.x;
  const float* Qb = Q + (size_t)b * kC * kN;
  const float* Pb = P + (size_t)b * kC * kN;

  __shared__ float sBp[kC];                 // proto channel means
  __shared__ float sAq[kC];                 // query channel means
  __shared__ float sMuQ[kN], sInvQ[kN];
  __shared__ float sMuP[kN], sInvP[kN];
  __shared__ float sW1[kN], sW2[kN];
  __shared__ float sSums[2];

  // ---- Phase A: channel means over the 49 pixels ----
  for (int c = t; c < kC; c += 128) {
    const float* pr = Pb + c * kN;
    const float* qr = Qb + c * kN;
    float sp = 0.f, sq = 0.f;
    for (int n = 0; n < kN; ++n) { sp += pr[n]; sq += qr[n]; }
    sBp[c] = sp * (1.0f / kN);
    sAq[c] = sq * (1.0f / kN);
  }
  __syncthreads();

  // ---- Phase B: per-pixel stats (49 active threads) ----
  if (t < kN) {
    float sq = 0.f, sqq = 0.f, wq = 0.f;
    float sp = 0.f, spp = 0.f, wp = 0.f;
    for (int c = 0; c < kC; ++c) {
      float qv = Qb[c * kN + t];
      float pv = Pb[c * kN + t];
      sq += qv; sqq += qv * qv; wq += qv * sBp[c];
      sp += pv; spp += pv * pv; wp += pv * sAq[c];
    }
    float mq = sq * (1.0f / kC);
    float mp = sp * (1.0f / kC);
    sMuQ[t] = mq;
    sMuP[t] = mp;
    // ||x - mu||^2 = sum x^2 - C*mu^2
    sInvQ[t] = 1.0f / (sqrtf(fmaxf(sqq - sq * mq, 0.0f)) + 1e-8f);
    sInvP[t] = 1.0f / (sqrtf(fmaxf(spp - sp * mp, 0.0f)) + 1e-8f);
    sW1[t] = fmaxf(wq, 0.0f) + 0.001f;
    sW2[t] = fmaxf(wp, 0.0f) + 0.001f;
  }
  __syncthreads();
  if (t == 0) {
    float s1 = 0.f, s2 = 0.f;
    for (int n = 0; n < kN; ++n) { s1 += sW1[n]; s2 += sW2[n]; }
    sSums[0] = s1; sSums[1] = s2;
  }
  __syncthreads();
  if (t < kN) {
    logr[b * kN + t] = __logf(sW1[t] / sSums[0]);
    logc[b * kN + t] = __logf(sW2[t] / sSums[1]);
  }

  // ---- Phase C: WMMA similarity matmul (all 128 lanes, EXEC all ones) ----
  const int wave = t >> 5;            // tile row 0..3
  const int lane = t & 31;
  const int half = lane >> 4;         // lane group 0/1 -> K pair select
  const int lm   = lane & 15;         // row/col within tile
  const int kk   = 2 * half;          // K offset within 4

  const int rA  = wave * 16 + lm;
  const int mA  = rA < kN ? rA : kN - 1;   // clamp: rows >=49 computed, discarded
  const float muA  = sMuQ[mA];
  const float invA = sInvQ[mA];

  int mB[4]; float muB[4], invB[4];
#pragma unroll
  for (int tc = 0; tc < 4; ++tc) {
    int rB = tc * 16 + lm;
    mB[tc]  = rB < kN ? rB : kN - 1;
    muB[tc] = sMuP[mB[tc]];
    invB[tc] = sInvP[mB[tc]];
  }

  v8f z = {0.f, 0.f, 0.f, 0.f, 0.f, 0.f, 0.f, 0.f};
  v8f acc[4] = {z, z, z, z};

#pragma unroll 2
  for (int k0 = 0; k0 < kC; k0 += 4) {
    const int ka = k0 + kk;
    v2f a;
    a.x = (Qb[ka * kN + mA] - muA) * invA;
    a.y = (Qb[(ka + 1) * kN + mA] - muA) * invA;
#pragma unroll
    for (int tc = 0; tc < 4; ++tc) {
      v2f bb;
      bb.x = (Pb[ka * kN + mB[tc]] - muB[tc]) * invB[tc];
      bb.y = (Pb[(ka + 1) * kN + mB[tc]] - muB[tc]) * invB[tc];
      // D = A(16x4,f32) * B(4x16,f32) + C
      acc[tc] = __builtin_amdgcn_wmma_f32_16x16x4_f32(
          false, a, false, bb, (short)0, acc[tc], false, false);
    }
  }

  // Unconditional padded store: rows/cols 49..63 are clamped duplicates,
  // simply ignored by kernel 2. No exec-mask juggling.
  float* simb = simw + (size_t)b * kNP * kNP;
#pragma unroll
  for (int tc = 0; tc < 4; ++tc) {
    const int col = tc * 16 + lm;
#pragma unroll
    for (int r = 0; r < 8; ++r) {
      const int row = wave * 16 + 8 * half + r;   // C/D layout: M = r + 8*(lane/16)
      simb[row * kNP + col] = acc[tc][r];
    }
  }
}

// ---------------------------------------------------------------------------
// Kernel 2: one wave32 per batch. Dual register layout (196 VGPRs):
//   m0[i]  = M[i][lane]       m1[i]  = M[i][lane+32]   (lane = column)
//   mt0[j] = M[lane][j]       mt1[j] = M[lane+32][j]   (lane = row)
// u-update: per-lane loop over j on mt*, v[j] broadcast via v_readlane.
// v-update: per-lane loop over i on m*,  u[i] broadcast via v_readlane.
// 4 rotating accumulators break the max/sum dependency chains (ILP for the
// ~1 resident wave per SIMD). No DS / shuffle traffic inside the 100-it loop.
// Out-of-range columns/rows pinned to -1e30 -> exp underflows to exactly 0.
// ---------------------------------------------------------------------------
__global__ __launch_bounds__(32)
void deepemd_sinkhorn_kernel(const float* __restrict__ simw,
                             const float* __restrict__ logr,
                             const float* __restrict__ logc,
                             float* __restrict__ out)
{
  const int b = blockIdx.x;
  const int lane = threadIdx.x;
  const float* Sb = simw + (size_t)b * kNP * kNP;
  const bool hi_ok = lane < (kN - 32);   // lanes 0..16 hold col/row 32..48

  float m0[kN], m1[kN], mt0[kN], mt1[kN];
#pragma unroll
  for (int i = 0; i < kN; ++i) {
    m0[i] = 20.0f * (Sb[i * kNP + lane] - 1.0f);          // M = -(1-sim)/0.05
    m1[i] = hi_ok ? 20.0f * (Sb[i * kNP + 32 + lane] - 1.0f) : kNegBig;
  }
#pragma unroll
  for (int j = 0; j < kN; ++j) {
    mt0[j] = 20.0f * (Sb[lane * kNP + j] - 1.0f);
    mt1[j] = hi_ok ? 20.0f * (Sb[(32 + lane) * kNP + j] - 1.0f) : kNegBig;
  }

  const float lr0 = logr[b * kN + lane];
  const float lr1 = hi_ok ? logr[b * kN + 32 + lane] : kNegBig;
  const float lc0 = logc[b * kN + lane];
  const float lc1 = hi_ok ? logc[b * kN + 32 + lane] : kNegBig;

  float v0 = 0.f, v1 = 0.f;   // lane = column
  float u0 = 0.f, u1 = 0.f;   // lane = row

#pragma unroll 1
  for (int it = 0; it < 100; ++it) {
    // ---- u[row] = log_r[row] - LSE_j(M[row, j] + v[j])  (lane = row) ----
    {
      float am0[4] = {-3.0e38f, -3.0e38f, -3.0e38f, -3.0e38f};
      float am1[4] = {-3.0e38f, -3.0e38f, -3.0e38f, -3.0e38f};
#pragma unroll
      for (int j = 0; j < kN; ++j) {
        float vj = (j < 32) ? lane_bcast(v0, j) : lane_bcast(v1, j - 32);
        am0[j & 3] = fmaxf(am0[j & 3], mt0[j] + vj);
        am1[j & 3] = fmaxf(am1[j & 3], mt1[j] + vj);
      }
      float tm0 = fmaxf(fmaxf(am0[0], am0[1]), fmaxf(am0[2], am0[3]));
      float tm1 = fmaxf(fmaxf(am1[0], am1[1]), fmaxf(am1[2], am1[3]));
      float s0[4] = {0.f, 0.f, 0.f, 0.f};
      float s1[4] = {0.f, 0.f, 0.f, 0.f};
#pragma unroll
      for (int j = 0; j < kN; ++j) {
        float vj = (j < 32) ? lane_bcast(v0, j) : lane_bcast(v1, j - 32);
        s0[j & 3] += __expf(mt0[j] + (vj - tm0));
        s1[j & 3] += __expf(mt1[j] + (vj - tm1));
      }
      u0 = lr0 - (tm0 + __logf((s0[0] + s0[1]) + (s0[2] + s0[3])));
      u1 = lr1 - (tm1 + __logf((s1[0] + s1[1]) + (s1[2] + s1[3])));
    }
    // ---- v[col] = log_c[col] - LSE_i(M[i, col] + u[i])  (lane = col) ----
    {
      float bm0[4] = {-3.0e38f, -3.0e38f, -3.0e38f, -3.0e38f};
      float bm1[4] = {-3.0e38f, -3.0e38f, -3.0e38f, -3.0e38f};
#pragma unroll
      for (int i = 0; i < kN; ++i) {
        float ui = (i < 32) ? lane_bcast(u0, i) : lane_bcast(u1, i - 32);
        bm0[i & 3] = fmaxf(bm0[i & 3], m0[i] + ui);
        bm1[i & 3] = fmaxf(bm1[i & 3], m1[i] + ui);
      }
      float mx0 = fmaxf(fmaxf(bm0[0], bm0[1]), fmaxf(bm0[2], bm0[3]));
      float mx1 = fmaxf(fmaxf(bm1[0], bm1[1]), fmaxf(bm1[2], bm1[3]));
      float s0[4] = {0.f, 0.f, 0.f, 0.f};
      float s1[4] = {0.f, 0.f, 0.f, 0.f};
#pragma unroll
      for (int i = 0; i < kN; ++i) {
        float ui = (i < 32) ? lane_bcast(u0, i) : lane_bcast(u1, i - 32);
        s0[i & 3] += __expf(m0[i] + (ui - mx0));
        s1[i & 3] += __expf(m1[i] + (ui - mx1));
      }
      v0 = lc0 - (mx0 + __logf((s0[0] + s0[1]) + (s0[2] + s0[3])));
      v1 = lc1 - (mx1 + __logf((s1[0] + s1[1]) + (s1[2] + s1[3])));
    }
  }

  // ---- logits = sum(exp(M+u+v) * sim) * T/N ;  sim = M/20 + 1 ----
  float acc0 = 0.f, acc1 = 0.f;
#pragma unroll
  for (int i = 0; i < kN; ++i) {
    float ui = (i < 32) ? lane_bcast(u0, i) : lane_bcast(u1, i - 32);
    acc0 += __expf(m0[i] + ui + v0) * (m0[i] * 0.05f + 1.0f);
    acc1 += __expf(m1[i] + ui + v1) * (m1[i] * 0.05f + 1.0f);  // exp(-1e30)=0
  }
  float acc = wave_sum32(acc0 + acc1);
  if (lane == 0) out[b] = acc * (12.5f / (float)kN);
}

// ---------------------------------------------------------------------------
extern "C" void kernel_launch(void* const* d_in, const int* in_sizes, int n_in,
                              void* d_out, int out_size, void* d_ws, size_t ws_size,
                              hipStream_t stream)
{
  const float* Q = (const float*)d_in[0];   // feature_map1 (query)
  const float* P = (const float*)d_in[1];   // feature_map2 (proto)
  const int B = in_sizes[0] / (kC * kN);    // 1024

  float* simw = (float*)d_ws;                        // B*64*64 f32 (~16.8 MB)
  float* logr = simw + (size_t)B * kNP * kNP;        // B*49
  float* logc = logr + (size_t)B * kN;               // B*49

  deepemd_sim_kernel<<<B, 128, 0, stream>>>(Q, P, simw, logr, logc);
  deepemd_sinkhorn_kernel<<<B, 32, 0, stream>>>(simw, logr, logc, (float*)d_out);
}